// CrystalGraphConv_7275674599728
// MI455X (gfx1250) — compile-verified
//
#include <hip/hip_runtime.h>

#define IN_DIM 128

typedef __attribute__((ext_vector_type(2))) float v2f;
typedef __attribute__((ext_vector_type(8))) float v8f;

// ---------------------------------------------------------------------------
// Kernel 1: node GEMMs via V_WMMA_F32_16X16X4_F32, B staged in LDS.
//   which==0 : h  = x @ W            + b   (also written straight to d_out)
//   which==1 : g1 = x @ Wg[0:128]
//   which==2 : g2 = x @ Wg[128:256]  + bg
// One wave computes a 16-row x 128-col strip (8 WMMA accumulator tiles).
// B is pre-swizzled into LDS as K-pair float2s so each lane's B fragment is a
// single ds_load_b64.
// ---------------------------------------------------------------------------
__global__ __launch_bounds__(128) void node_gemm_kernel(
    const float* __restrict__ x,
    const float* __restrict__ W,
    const float* __restrict__ b,
    const float* __restrict__ Wg,
    const float* __restrict__ bg,
    float* __restrict__ ws,
    float* __restrict__ out,
    int n_nodes, int n_row_tiles)
{
    __shared__ float2 ldsB[64 * 128];   // 64 KB: B as {B[2k2][n], B[2k2+1][n]}

    const int lane = threadIdx.x & 31;
    const int wv   = threadIdx.x >> 5;
    const int hlf  = lane >> 4;    // which 16-lane half of the wave
    const int idx  = lane & 15;    // position within half
    const int which = blockIdx.y;

    const float* Bmat = (which == 0) ? W : (Wg + (which == 2 ? IN_DIM * IN_DIM : 0));
    const float* bias = (which == 0) ? b : ((which == 2) ? bg : nullptr);
    float* Out = ws + (size_t)which * (size_t)n_nodes * IN_DIM;

    // ---- cooperative B staging (pre-swizzled K-pairs), all threads ----
    for (int i = threadIdx.x; i < 64 * 128; i += 128) {
        const int k2 = i >> 7;      // K-pair index 0..63
        const int n  = i & 127;
        float2 v;
        v.x = Bmat[(size_t)(2 * k2)     * IN_DIM + n];
        v.y = Bmat[(size_t)(2 * k2 + 1) * IN_DIM + n];
        ldsB[i] = v;
    }
    __syncthreads();

    const int rowTile = blockIdx.x * 4 + wv;   // wave-uniform -> safe to skip
    if (rowTile >= n_row_tiles) return;
    const int row0 = rowTile * 16;

    v8f acc[8];
    #pragma unroll
    for (int t = 0; t < 8; ++t) { v8f z = {}; acc[t] = z; }

    // A: lane (hlf,idx) holds row (row0+idx), K elements {k+2*hlf, k+2*hlf+1}
    int arow_i = row0 + idx; if (arow_i >= n_nodes) arow_i = n_nodes - 1;
    const float* arow = x + (size_t)arow_i * IN_DIM + 2 * hlf;

    for (int k = 0; k < IN_DIM; k += 4) {
        v2f a = *(const v2f*)(arow + k);                 // b64 global load
        const float2* bbase = ldsB + (size_t)((k >> 1) + hlf) * 128 + idx;
        #pragma unroll
        for (int nt = 0; nt < 8; ++nt) {
            float2 bp = bbase[nt * 16];                  // ds_load_b64
            v2f bb; bb.x = bp.x; bb.y = bp.y;            // (K=k+2h, K=k+2h+1)
            acc[nt] = __builtin_amdgcn_wmma_f32_16x16x4_f32(
                false, a, false, bb, (short)0, acc[nt], false, false);
        }
    }

    // C/D layout: VGPR j -> row (j + 8*hlf), lane idx -> col
    #pragma unroll
    for (int nt = 0; nt < 8; ++nt) {
        const int n  = nt * 16 + idx;
        const float bv = bias ? bias[n] : 0.0f;
        #pragma unroll
        for (int j = 0; j < 8; ++j) {
            const int r = row0 + j + 8 * hlf;
            if (r < n_nodes) {
                const float val = acc[nt][j] + bv;
                Out[(size_t)r * IN_DIM + n] = val;
                if (which == 0)                       // fused self term
                    out[(size_t)r * IN_DIM + n] = val;
            }
        }
    }
}

// ---------------------------------------------------------------------------
// Kernel 2: per-edge gather + gated message + f32 atomic scatter.
// 32 lanes per edge, float4 per lane. All node arrays are L2-resident
// (4 x 25.6 MB << 192 MB L2).
// ---------------------------------------------------------------------------
__global__ __launch_bounds__(256) void edge_kernel(
    const long long* __restrict__ ei,
    const float* __restrict__ g1,
    const float* __restrict__ g2,
    const float* __restrict__ h,
    float* __restrict__ out,
    int n_edges)
{
    long long tid = (long long)blockIdx.x * blockDim.x + threadIdx.x;
    int e = (int)(tid >> 5);
    if (e >= n_edges) return;
    int lane = (int)(tid & 31);

    const long long r = ei[e];
    const long long c = ei[(long long)n_edges + e];

    const float4 gr = *(const float4*)(g1 + (size_t)r * IN_DIM + lane * 4);
    const float4 gc = *(const float4*)(g2 + (size_t)c * IN_DIM + lane * 4);
    const float4 hv = *(const float4*)(h  + (size_t)c * IN_DIM + lane * 4);
    float* dst = out + (size_t)r * IN_DIM + lane * 4;

    float zx = gr.x + gc.x, zy = gr.y + gc.y, zz = gr.z + gc.z, zw = gr.w + gc.w;
    atomicAdd(dst + 0, hv.x / (1.0f + __expf(-zx)));
    atomicAdd(dst + 1, hv.y / (1.0f + __expf(-zy)));
    atomicAdd(dst + 2, hv.z / (1.0f + __expf(-zz)));
    atomicAdd(dst + 3, hv.w / (1.0f + __expf(-zw)));
}

// ---------------------------------------------------------------------------
extern "C" void kernel_launch(void* const* d_in, const int* in_sizes, int n_in,
                              void* d_out, int out_size, void* d_ws, size_t ws_size,
                              hipStream_t stream)
{
    const float*     x  = (const float*)d_in[0];
    const float*     W  = (const float*)d_in[1];
    const float*     b  = (const float*)d_in[2];
    const float*     Wg = (const float*)d_in[3];
    const float*     bg = (const float*)d_in[4];
    const long long* ei = (const long long*)d_in[5];
    float* out = (float*)d_out;
    float* ws  = (float*)d_ws;

    const int N = in_sizes[0] / IN_DIM;
    const int E = in_sizes[5] / 2;
    const int tiles = (N + 15) / 16;

    // ws layout: h | g1 | g2  (each N*128 floats)
    float* h  = ws;
    float* g1 = ws + (size_t)N * IN_DIM;
    float* g2 = ws + 2 * (size_t)N * IN_DIM;

    dim3 gemm_grid((tiles + 3) / 4, 3);
    node_gemm_kernel<<<gemm_grid, 128, 0, stream>>>(x, W, b, Wg, bg, ws, out, N, tiles);

    const long long work = (long long)E * 32;
    edge_kernel<<<(unsigned)((work + 255) / 256), 256, 0, stream>>>(ei, g1, g2, h, out, E);
}